// N3Aggregation2D_34943853920739
// MI455X (gfx1250) — compile-verified
//
#include <hip/hip_runtime.h>
#include <math.h>
#include <stdint.h>

// ---------------- problem constants (from reference) ----------------
#define PS      10          // patch size
#define K_NBR   7           // top-k neighbors
#define WSZ     29          // search window
#define RAD     14          // WSZ/2
#define HH      98          // h = w = 96 + 2 (pad1)
#define HP      107         // patch-padded extent = HH + PS - 1
#define NP      (HP * HP)   // 11449
#define IMG     96
#define CE      8           // embed channels
#define CX      16          // x channels
#define NQ      (HH * HH)   // 9604 queries

typedef __attribute__((ext_vector_type(2))) float v2f;
typedef __attribute__((ext_vector_type(8))) float v8f;

// LDS partition (float indices) for the search kernel (dynamic LDS, base = 0)
#define S_YE    0                   // [8][25][26]        = 5200
#define S_XE    5200                // [8][53][54]        = 22896
#define S_SQ    28096               // [32 cols][28 rows] = 896  (column-major, padded)
#define S_P     28992               // [16][32]           = 512  (row-major, padded)
#define S_D     29504               // [16][16]           = 256
#define S_TOT   29760               // floats -> 119040 bytes (< 320KB/WGP)
#define S_ZERO  28992               // zero-fill s_ye + s_xe + s_sq (incl. pads)

// ==================================================================
// Kernel 1: build zero-padded embeddings: pad1 (+1) then patch-pad
// (+ADJ=5) => offset 6 into a 107x107 frame, zeros elsewhere.
// ==================================================================
__global__ __launch_bounds__(256) void prep_kernel(
    const float* __restrict__ xe, const float* __restrict__ ye,
    float* __restrict__ xep, float* __restrict__ yep) {
  int e = blockIdx.x * 256 + threadIdx.x;
  if (e >= CE * NP) return;
  int c   = e / NP;
  int rem = e - c * NP;
  int u   = rem / HP;
  int v   = rem - u * HP;
  int r = u - 6, q = v - 6;
  bool inb = (r >= 0) && (r < IMG) && (q >= 0) && (q < IMG);
  int src = c * (IMG * IMG) + r * IMG + q;
  xep[e] = inb ? xe[src] : 0.0f;
  yep[e] = inb ? ye[src] : 0.0f;
}

// per-lane async global->LDS copy of one dword (CDNA5 ASYNCcnt path)
__device__ __forceinline__ void async_g2l_b32(unsigned lds_byte_off,
                                              const float* gptr) {
  unsigned long long ga = (unsigned long long)(uintptr_t)gptr;
  asm volatile("global_load_async_to_lds_b32 %0, %1, off"
               :: "v"(lds_byte_off), "v"(ga)
               : "memory");
}

// ==================================================================
// Kernel 2: patch search. One block = 16x16 query tile.
//   per displacement d=(dy,dx):
//     sq[a][b] = sum_c (yep - xep_shift)^2 over 25x25 support
//     dist(16x16) = M * sq * M^T   via V_WMMA_F32_16X16X4_F32
//     per-thread top-7 insertion
//   then softmax weights + neighbor flat indices.
// ==================================================================
__global__ __launch_bounds__(256) void search_kernel(
    const float* __restrict__ xep, const float* __restrict__ yep,
    float* __restrict__ wgt, int* __restrict__ nbr) {
  extern __shared__ float smem[];
  float* s_ye = smem + S_YE;
  float* s_xe = smem + S_XE;
  float* s_sq = smem + S_SQ;   // column-major: [col][row], 28 rows per col
  float* s_P  = smem + S_P;    // row-major:    [m][32]
  float* s_d  = smem + S_D;

  const int tid = threadIdx.x;
  const int qi0 = (blockIdx.x / 7) * 16;
  const int qj0 = (blockIdx.x % 7) * 16;

  // ---- zero-fill staging regions + sq pads (provides zero-pad semantics) ----
  for (int t = tid; t < S_ZERO; t += 256) smem[t] = 0.0f;
  __syncthreads();  // ds stores complete (dscnt) before async writes land

  // ---- async-DMA stage ye tile: rows/cols [qi0, qi0+24] ----
  for (int t = tid; t < CE * 25 * 25; t += 256) {
    int c = t / 625, rr = t - c * 625;
    int a = rr / 25, b = rr - a * 25;
    int gr = qi0 + a, gc = qj0 + b;
    if (gr < HP && gc < HP)
      async_g2l_b32((S_YE + c * 650 + a * 26 + b) * 4u,
                    yep + c * NP + gr * HP + gc);
  }
  // ---- async-DMA stage xe search region: rows/cols [qi0-14, qi0+38] ----
  for (int t = tid; t < CE * 53 * 53; t += 256) {
    int c = t / 2809, rr = t - c * 2809;
    int a = rr / 53, b = rr - a * 53;
    int gr = qi0 - RAD + a, gc = qj0 - RAD + b;
    if (gr >= 0 && gr < HP && gc >= 0 && gc < HP)
      async_g2l_b32((S_XE + c * 2862 + a * 54 + b) * 4u,
                    xep + c * NP + gr * HP + gc);
  }
  asm volatile("s_wait_asynccnt 0x0" ::: "memory");
  __syncthreads();

  const int ty = tid >> 4, tx = tid & 15;
  const int qi = qi0 + ty, qj = qj0 + tx;
  const int lane = tid & 31;
  const int lo = lane & 15, hi = lane >> 4;
  const bool w0 = (tid < 32);

  // loop-invariant banded matrix M[m][k] = (k-m in [0,10)), zero past col 24.
  // Serves as A in step A and as M^T (=B) in step B. ka = 4*kk + 2*hi.
  v2f Mv[7];
#pragma unroll
  for (int kk = 0; kk < 7; ++kk) {
    int ka = kk * 4 + 2 * hi;
    Mv[kk].x = (ka >= lo && ka < lo + PS && ka < 25) ? 1.0f : 0.0f;
    Mv[kk].y = (ka + 1 >= lo && ka + 1 < lo + PS && ka + 1 < 25) ? 1.0f : 0.0f;
  }

  float bd[K_NBR];
  int bi[K_NBR];
#pragma unroll
  for (int k = 0; k < K_NBR; ++k) { bd[k] = 3.0e38f; bi[k] = 0; }

  for (int d = 0; d < WSZ * WSZ; ++d) {
    const int dy = d / WSZ - RAD;
    const int dx = d % WSZ - RAD;

    // squared-diff map over the 25x25 box-sum support (column-major store)
    for (int t = tid; t < 625; t += 256) {
      int a = t / 25, b = t - (t / 25) * 25;
      const float* yp = s_ye + a * 26 + b;
      const float* xp = s_xe + (a + RAD + dy) * 54 + (b + RAD + dx);
      float s = 0.0f;
#pragma unroll
      for (int c = 0; c < CE; ++c) {
        float df = yp[c * 650] - xp[c * 2862];
        s += df * df;
      }
      s_sq[b * 28 + a] = s;
    }
    __syncthreads();

    if (w0) {
      // -------- step A: P(16x25) = M(16x25) * S(25x25) --------
#pragma unroll
      for (int nt = 0; nt < 2; ++nt) {
        v8f acc = {0.f, 0.f, 0.f, 0.f, 0.f, 0.f, 0.f, 0.f};
        const int col = nt * 16 + lo;                 // 0..31 (pads are zero)
#pragma unroll
        for (int kk = 0; kk < 7; ++kk) {
          const int ka = kk * 4 + 2 * hi;
          // B = S slice: rows (ka, ka+1) of column 'col' -> contiguous b64
          v2f B = *(const v2f*)(s_sq + col * 28 + ka);
          acc = __builtin_amdgcn_wmma_f32_16x16x4_f32(
              false, Mv[kk], false, B, (short)0, acc, false, false);
        }
        // unconditional store; cols >= 25 are exactly 0 (zero-padded S)
#pragma unroll
        for (int j = 0; j < 8; ++j) s_P[(j + 8 * hi) * 32 + col] = acc[j];
      }
      asm volatile("s_wait_dscnt 0x0" ::: "memory");  // wave-local P handoff
      // -------- step B: D(16x16) = P(16x25) * M^T(25x16) --------
      v8f acc2 = {0.f, 0.f, 0.f, 0.f, 0.f, 0.f, 0.f, 0.f};
#pragma unroll
      for (int kk = 0; kk < 7; ++kk) {
        const int ka = kk * 4 + 2 * hi;
        // A = P: elements (lo, ka), (lo, ka+1) -> contiguous b64
        v2f A = *(const v2f*)(s_P + lo * 32 + ka);
        acc2 = __builtin_amdgcn_wmma_f32_16x16x4_f32(
            false, A, false, Mv[kk], (short)0, acc2, false, false);
      }
#pragma unroll
      for (int j = 0; j < 8; ++j) s_d[(j + 8 * hi) * 16 + lo] = acc2[j];
    }
    __syncthreads();

    // top-k maintenance (matches top_k tie-breaking: earlier d wins ties)
    float dist = s_d[ty * 16 + tx];
    const int ni = qi + dy, nj = qj + dx;
    const bool valid = (ni >= 0) && (ni < HH) && (nj >= 0) && (nj < HH) &&
                       !((dy == 0) && (dx == 0));
    if (!valid) dist = 1.0e10f;
    if (dist < bd[K_NBR - 1]) {
      int p = K_NBR - 1;
      while (p > 0 && bd[p - 1] > dist) {
        bd[p] = bd[p - 1];
        bi[p] = bi[p - 1];
        --p;
      }
      bd[p] = dist;
      bi[p] = d;
    }
    __syncthreads();
  }

  if (qi < HH && qj < HH) {
    const int q = qi * HH + qj;
    float e[K_NBR], s = 0.0f;
    const float m0 = bd[0];
#pragma unroll
    for (int k = 0; k < K_NBR; ++k) { e[k] = expf(-(bd[k] - m0)); s += e[k]; }
    const float inv = 1.0f / s;
#pragma unroll
    for (int k = 0; k < K_NBR; ++k) {
      wgt[q * K_NBR + k] = e[k] * inv;
      const int dd = bi[k];
      const int dy = dd / WSZ - RAD, dx = dd % WSZ - RAD;
      nbr[q * K_NBR + k] = (qi + dy) * HH + (qj + dx);
    }
  }
}

// ==================================================================
// Kernel 3: gather-form fold + output assembly.
//   out[c]      = x[c]                               (y cropped)
//   out[16 + c] = vid[c]/zvid - x[c]                 (z cropped)
// vid gathered over covering queries -> deterministic, no atomics.
// ==================================================================
__device__ __forceinline__ int imin_(int a, int b) { return a < b ? a : b; }
__device__ __forceinline__ int imax_(int a, int b) { return a > b ? a : b; }

__global__ __launch_bounds__(256) void aggregate_kernel(
    const float* __restrict__ x, const float* __restrict__ wgt,
    const int* __restrict__ nbr, float* __restrict__ out) {
  const int p = blockIdx.x * 256 + threadIdx.x;
  if (p >= IMG * IMG) return;
  const int i = p / IMG, j = p - (p / IMG) * IMG;
  const int a = i + 6, b = j + 6;  // vid coordinates (crop 1 + ADJ 5)

  float acc[CX];
#pragma unroll
  for (int c = 0; c < CX; ++c) acc[c] = 0.0f;

  for (int oy = 0; oy < PS; ++oy) {
    const int qi = a - oy;
    if (qi < 0 || qi >= HH) continue;
    for (int ox = 0; ox < PS; ++ox) {
      const int qj = b - ox;
      if (qj < 0 || qj >= HH) continue;
      const int q = qi * HH + qj;
#pragma unroll
      for (int k = 0; k < K_NBR; ++k) {
        const float w = wgt[q * K_NBR + k];
        const int n  = nbr[q * K_NBR + k];
        const int ni = n / HH, nj = n - (n / HH) * HH;
        const int xr = ni + oy - 6, xc = nj + ox - 6;  // into original x
        if (xr >= 0 && xr < IMG && xc >= 0 && xc < IMG) {
          const float* xp = x + xr * IMG + xc;
#pragma unroll
          for (int c = 0; c < CX; ++c) acc[c] += w * xp[c * IMG * IMG];
        }
      }
    }
  }

  const float cr = (float)(imin_(a, HH - 1) - imax_(a - PS + 1, 0) + 1);
  const float cb = (float)(imin_(b, HH - 1) - imax_(b - PS + 1, 0) + 1);
  const float inv = 1.0f / (cr * cb);
#pragma unroll
  for (int c = 0; c < CX; ++c) {
    const float xv = x[c * IMG * IMG + i * IMG + j];
    out[c * IMG * IMG + i * IMG + j] = xv;                        // y half
    out[(CX + c) * IMG * IMG + i * IMG + j] = acc[c] * inv - xv;  // z half
  }
}

// ==================================================================
extern "C" void kernel_launch(void* const* d_in, const int* in_sizes, int n_in,
                              void* d_out, int out_size, void* d_ws,
                              size_t ws_size, hipStream_t stream) {
  const float* x  = (const float*)d_in[0];
  const float* xe = (const float*)d_in[1];
  const float* ye = (const float*)d_in[2];

  float* ws  = (float*)d_ws;
  const int NPE = CE * NP;            // 91592 floats per padded embedding
  float* yep = ws;
  float* xep = ws + NPE;
  float* wgt = ws + 2 * NPE;          // NQ*7 floats
  int*   nbr = (int*)(ws + 2 * NPE + NQ * K_NBR);

  prep_kernel<<<(NPE + 255) / 256, 256, 0, stream>>>(xe, ye, xep, yep);
  search_kernel<<<49, 256, S_TOT * sizeof(float), stream>>>(xep, yep, wgt, nbr);
  aggregate_kernel<<<(IMG * IMG + 255) / 256, 256, 0, stream>>>(
      x, wgt, nbr, (float*)d_out);
}